// SVAE_LDS_24283745092108
// MI455X (gfx1250) — compile-verified
//
#include <hip/hip_runtime.h>
#include <hip/hip_bf16.h>
#include <math.h>

#define BB   256
#define TT   512
#define OBS  6
#define DD   12
#define HID  20
#define NW   4          // waves per block in filter kernel
#define S16  256        // 16x16 padded matrix (floats)

typedef __attribute__((ext_vector_type(2))) float v2f;
typedef __attribute__((ext_vector_type(8))) float v8f;

// ---------------------------------------------------------------------------
// 16x16 (K=16) f32 matmul on LDS tiles via 4 chained V_WMMA_F32_16X16X4_F32.
// A,B,D are 16x16 row-major (stride 16) LDS tiles; only 12x12 is meaningful,
// padding must be zero in the operands. TA/TB: use operand transposed.
// D = A(^T) x B(^T) [+ addC if ADD].  Called by all 32 lanes of every wave
// (EXEC all-ones at every call site, per WMMA restriction).
// ---------------------------------------------------------------------------
template <bool TA, bool TB, bool ADD>
__device__ __forceinline__ void mm16(const float* A_, const float* B_,
                                     float* D_, const float* addC, int lane) {
  __syncthreads();
  const int hf = lane >> 4;      // which half of the wave
  const int lm = lane & 15;      // M for A, N for B/C/D
  v8f c = {};
#pragma unroll
  for (int k0 = 0; k0 < 16; k0 += 4) {
    const int ka = k0 + hf * 2;  // K pair handled by this half
    v2f a, b;
    if (TA) { a.x = A_[ka * 16 + lm];      a.y = A_[(ka + 1) * 16 + lm]; }
    else    { a.x = A_[lm * 16 + ka];      a.y = A_[lm * 16 + ka + 1];   }
    if (TB) { b.x = B_[lm * 16 + ka];      b.y = B_[lm * 16 + ka + 1];   }
    else    { b.x = B_[ka * 16 + lm];      b.y = B_[(ka + 1) * 16 + lm]; }
    c = __builtin_amdgcn_wmma_f32_16x16x4_f32(false, a, false, b,
                                              (short)0, c, false, false);
  }
  __syncthreads();
#pragma unroll
  for (int r = 0; r < 8; ++r) {
    const int idx = (r + hf * 8) * 16 + lm;   // row = r + 8*half, col = lm
    if (ADD) D_[idx] = c[r] + addC[idx];
    else     D_[idx] = c[r];
  }
  __syncthreads();
}

// Cooperative in-place Cholesky of the 12x12 block of a 16-stride LDS tile.
// Lower triangle gets L; upper triangle left as-is (never used as operand).
__device__ __forceinline__ void chol12(float* M, int lane) {
  for (int k = 0; k < DD; ++k) {
    __syncthreads();
    if (lane == 0) M[k * 16 + k] = sqrtf(fmaxf(M[k * 16 + k], 1e-30f));
    __syncthreads();
    if (lane > k && lane < DD) M[lane * 16 + k] /= M[k * 16 + k];
    __syncthreads();
    if (lane > k && lane < DD) {
      const float lik = M[lane * 16 + k];
      for (int j = k + 1; j <= lane; ++j) M[lane * 16 + j] -= lik * M[j * 16 + k];
    }
  }
  __syncthreads();
}

// X = L^-1 (lower triangular), column-parallel (lane j owns column j).
// X fully zeroed first so it is a clean padded WMMA operand.
__device__ __forceinline__ void linv12(const float* L, float* X, int lane) {
  __syncthreads();
  for (int i = lane; i < S16; i += 32) X[i] = 0.0f;
  __syncthreads();
  if (lane < DD) {
    const int j = lane;
    float x[DD];
    for (int i = 0; i < DD; ++i) x[i] = 0.0f;
    x[j] = 1.0f / L[j * 16 + j];
    for (int i = j + 1; i < DD; ++i) {
      float s = 0.0f;
      for (int m = j; m < i; ++m) s += L[i * 16 + m] * x[m];
      x[i] = -s / L[i * 16 + i];
    }
    for (int i = j; i < DD; ++i) X[i * 16 + j] = x[i];
  }
  __syncthreads();
}

// ---------------------------------------------------------------------------
// Encoder: one thread per (b,t). h=relu(x W1+b1); xhat=h Wm+bm; sg=exp(.5(h Wl+bl))
// ---------------------------------------------------------------------------
__global__ void svae_enc_kernel(const float* __restrict__ x,
                                const float* __restrict__ w1, const float* __restrict__ b1,
                                const float* __restrict__ wm, const float* __restrict__ bm,
                                const float* __restrict__ wl, const float* __restrict__ bl,
                                float* __restrict__ xh, float* __restrict__ sg) {
  const int idx = blockIdx.x * blockDim.x + threadIdx.x;
  if (idx >= BB * TT) return;
  float xi[OBS];
#pragma unroll
  for (int i = 0; i < OBS; ++i) xi[i] = x[(size_t)idx * OBS + i];
  float h[HID];
#pragma unroll
  for (int j = 0; j < HID; ++j) {
    float s = b1[j];
#pragma unroll
    for (int i = 0; i < OBS; ++i) s += xi[i] * w1[i * HID + j];
    h[j] = fmaxf(s, 0.0f);
  }
#pragma unroll
  for (int d = 0; d < DD; ++d) {
    float mu = bm[d], lv = bl[d];
#pragma unroll
    for (int j = 0; j < HID; ++j) { mu += h[j] * wm[j * DD + d]; lv += h[j] * wl[j * DD + d]; }
    xh[(size_t)idx * DD + d] = mu;
    sg[(size_t)idx * DD + d] = expf(0.5f * lv);
  }
}

// ---------------------------------------------------------------------------
// Build Q = L L^T from packed lower-tri Q_param (exp on diagonal). One block.
// ---------------------------------------------------------------------------
__global__ void svae_prep_kernel(const float* __restrict__ qp, float* __restrict__ Qout) {
  __shared__ float L[DD][DD];
  const int tid = threadIdx.x;
  if (tid < DD * DD) {
    const int i = tid / DD, j = tid % DD;
    float v = 0.0f;
    if (j <= i) {
      const float p = qp[i * (i + 1) / 2 + j];
      v = (i == j) ? expf(p) : p;
    }
    L[i][j] = v;
  }
  __syncthreads();
  if (tid < DD * DD) {
    const int i = tid / DD, j = tid % DD;
    float s = 0.0f;
    for (int k = 0; k < DD; ++k) s += L[i][k] * L[j][k];
    Qout[i * DD + j] = s;
  }
}

// ---------------------------------------------------------------------------
// Filter + smoother + sampler + decoder. One wave32 per batch chain.
// ---------------------------------------------------------------------------
__global__ void __launch_bounds__(NW * 32)
svae_filter_kernel(const float* __restrict__ xh_g, const float* __restrict__ sg_g,
                   const float* __restrict__ Qg,   const int*   __restrict__ msk_g,
                   const float* __restrict__ eps_g,
                   const float* __restrict__ Ag,   const float* __restrict__ bg,
                   const float* __restrict__ dwg,  const float* __restrict__ dbg,
                   float* __restrict__ o_xr, float* __restrict__ o_z,
                   float* __restrict__ o_qm, float* __restrict__ o_qc,
                   float* __restrict__ o_ll) {
  __shared__ float sA[S16], sAt[S16], sQ[S16];
  __shared__ float sb[16], sdecw[DD * 8], sdecb[8];
  __shared__ float wbuf[NW][9 * S16];
  __shared__ float wvec[NW][96];

  const int lane = threadIdx.x & 31;
  const int wid  = threadIdx.x >> 5;
  const int bb   = blockIdx.x * NW + wid;

  // shared constants
  for (int i = threadIdx.x; i < S16; i += blockDim.x) {
    const int r = i >> 4, c = i & 15;
    const bool ok = (r < DD) && (c < DD);
    sA[i]  = ok ? Ag[r * DD + c] : 0.0f;
    sAt[i] = ok ? Ag[c * DD + r] : 0.0f;
    sQ[i]  = ok ? Qg[r * DD + c] : 0.0f;
  }
  if (threadIdx.x < 16) sb[threadIdx.x] = (threadIdx.x < DD) ? bg[threadIdx.x] : 0.0f;
  for (int i = threadIdx.x; i < DD * 8; i += blockDim.x) {
    const int r = i >> 3, c = i & 7;
    sdecw[i] = (c < OBS) ? dwg[r * OBS + c] : 0.0f;
  }
  if (threadIdx.x < 8) sdecb[threadIdx.x] = (threadIdx.x < OBS) ? dbg[threadIdx.x] : 0.0f;

  float* MP  = wbuf[wid] + 0 * S16;   // P carry / Pf(t)
  float* MPP = wbuf[wid] + 1 * S16;   // P_pred
  float* MT1 = wbuf[wid] + 2 * S16;
  float* MSL = wbuf[wid] + 3 * S16;   // S -> chol L (reused)
  float* MLI = wbuf[wid] + 4 * S16;   // L^-1
  float* MSI = wbuf[wid] + 5 * S16;   // S^-1 / P_pred^-1
  float* MK  = wbuf[wid] + 6 * S16;   // K / G
  float* MT2 = wbuf[wid] + 7 * S16;
  float* MPS = wbuf[wid] + 8 * S16;   // smoothed P carry
  float* vm  = wvec[wid] + 0;         // m carry / ms carry
  float* vmp = wvec[wid] + 16;        // m_pred
  float* vxh = wvec[wid] + 32;        // xhat / mf(t)
  float* vsg = wvec[wid] + 48;        // sigma
  float* vin = wvec[wid] + 64;        // innovation / (ms - m_pred)
  float* vz  = wvec[wid] + 80;        // sampled z

  // init carry: m=0, P=I (padded)
  for (int i = lane; i < S16; i += 32) {
    const int r = i >> 4, c = i & 15;
    MP[i] = (r == c && r < DD) ? 1.0f : 0.0f;
  }
  if (lane < 16) vm[lane] = 0.0f;
  __syncthreads();

  const float LOG2PI = 1.8378770664093453f;
  float llsum = 0.0f;

  // ============================ forward filter ============================
  for (int t = 0; t < TT; ++t) {
    const size_t bt = (size_t)bb * TT + t;
    // overlap next step's HBM fetch with this step's matrix work
    if (t + 1 < TT && lane == 0) {
      __builtin_prefetch(xh_g + (bt + 1) * DD, 0, 3);
      __builtin_prefetch(sg_g + (bt + 1) * DD, 0, 3);
      __builtin_prefetch(msk_g + bt + 1, 0, 3);
    }
    if (lane < DD) {
      vxh[lane] = xh_g[bt * DD + lane];
      vsg[lane] = sg_g[bt * DD + lane];
    }
    const int mski = msk_g[bt];
    __syncthreads();

    if (lane < DD) {                       // m_pred = A m + b
      float s = sb[lane];
      for (int j = 0; j < DD; ++j) s += sA[lane * 16 + j] * vm[j];
      vmp[lane] = s;
    }
    mm16<false, false, false>(sA, MP, MT1, nullptr, lane);   // A P
    mm16<false, false, true >(MT1, sAt, MPP, sQ, lane);      // A P A^T + Q

    for (int i = lane; i < S16; i += 32) {                   // S = P_pred + diag(sg)
      const int r = i >> 4, c = i & 15;
      MSL[i] = MPP[i] + ((r == c && r < DD) ? vsg[r] : 0.0f);
    }
    chol12(MSL, lane);
    linv12(MSL, MLI, lane);
    mm16<true,  false, false>(MLI, MLI, MSI, nullptr, lane); // Sinv = Linv^T Linv
    mm16<false, false, false>(MPP, MSI, MK, nullptr, lane);  // K = P_pred Sinv

    if (lane < DD) vin[lane] = vxh[lane] - vmp[lane];
    __syncthreads();
    if (lane < DD) {                                         // m update + select
      float s = vmp[lane];
      for (int j = 0; j < DD; ++j) s += MK[lane * 16 + j] * vin[j];
      const float mf = mski ? vmp[lane] : s;
      vm[lane] = mf;
      o_qm[bt * DD + lane] = mf;
    }
    mm16<false, false, false>(MK, MPP, MT2, nullptr, lane);  // K P_pred
    for (int i = lane; i < S16; i += 32)                     // P update + select
      MP[i] = mski ? MPP[i] : (MPP[i] - MT2[i]);
    __syncthreads();
    for (int e = lane; e < DD * DD; e += 32)
      o_qc[bt * (DD * DD) + e] = MP[(e / DD) * 16 + (e % DD)];

    if (lane == 0 && !mski) {                                // log-likelihood
      float ld = 0.0f;
      for (int k = 0; k < DD; ++k) ld += logf(MSL[k * 16 + k]);
      ld *= 2.0f;
      float maha = 0.0f;
      for (int i = 0; i < DD; ++i) {
        const float vi = vin[i];
        for (int j = 0; j < DD; ++j) maha += vi * MSI[i * 16 + j] * vin[j];
      }
      llsum += -0.5f * ((float)DD * LOG2PI + ld + maha);
    }
    __syncthreads();
  }
  if (lane == 0) o_ll[bb] = llsum;

  // ============================ RTS smoother ==============================
  __syncthreads();
  for (int i = lane; i < S16; i += 32) MPS[i] = MP[i];  // Ps carry = Pf[T-1]
  __syncthreads();

  for (int t = TT - 2; t >= 0; --t) {
    const size_t bt = (size_t)bb * TT + t;
    if (t > 0 && lane < 3) {   // prefetch next (t-1) filtered state, 144 floats
      __builtin_prefetch(o_qc + (bt - 1) * (DD * DD) + lane * 48, 0, 3);
      if (lane == 0) __builtin_prefetch(o_qm + (bt - 1) * DD, 0, 3);
    }
    if (lane < DD) vxh[lane] = o_qm[bt * DD + lane];    // mf(t)
    for (int i = lane; i < S16; i += 32) {              // Pf(t), padded
      const int r = i >> 4, c = i & 15;
      MP[i] = (r < DD && c < DD) ? o_qc[bt * (DD * DD) + r * DD + c] : 0.0f;
    }
    __syncthreads();

    if (lane < DD) {                                    // m_pred = A mf + b
      float s = sb[lane];
      for (int j = 0; j < DD; ++j) s += sA[lane * 16 + j] * vxh[j];
      vmp[lane] = s;
    }
    mm16<false, false, false>(sA, MP, MT1, nullptr, lane);
    mm16<false, false, true >(MT1, sAt, MPP, sQ, lane); // P_pred

    for (int i = lane; i < S16; i += 32) MSL[i] = MPP[i];
    chol12(MSL, lane);
    linv12(MSL, MLI, lane);
    mm16<true,  false, false>(MLI, MLI, MSI, nullptr, lane); // P_pred^-1
    mm16<false, false, false>(MP, sAt, MT1, nullptr, lane);  // Pf A^T
    mm16<false, false, false>(MT1, MSI, MK, nullptr, lane);  // G

    if (lane < DD) vin[lane] = vm[lane] - vmp[lane];    // ms - m_pred
    __syncthreads();
    if (lane < DD) {                                    // m_new
      float s = vxh[lane];
      for (int j = 0; j < DD; ++j) s += MK[lane * 16 + j] * vin[j];
      vm[lane] = s;
      o_qm[bt * DD + lane] = s;
    }
    for (int i = lane; i < S16; i += 32) MT2[i] = MPS[i] - MPP[i];  // Ps - P_pred
    mm16<false, false, false>(MK, MT2, MT1, nullptr, lane);  // G (Ps - P_pred)
    mm16<false, true,  false>(MT1, MK, MSL, nullptr, lane);  // ... G^T
    for (int i = lane; i < S16; i += 32) MPS[i] = MP[i] + MSL[i];   // P_new
    __syncthreads();
    for (int e = lane; e < DD * DD; e += 32)
      o_qc[bt * (DD * DD) + e] = MPS[(e / DD) * 16 + (e % DD)];
    __syncthreads();
  }

  // ===================== sample z, decode x_recon =========================
  for (int t = 0; t < TT; ++t) {
    const size_t bt = (size_t)bb * TT + t;
    if (t + 1 < TT && lane < 3) {
      __builtin_prefetch(o_qc + (bt + 1) * (DD * DD) + lane * 48, 0, 3);
      if (lane == 0) __builtin_prefetch(eps_g + (bt + 1) * DD, 0, 3);
    }
    for (int i = lane; i < S16; i += 32) {
      const int r = i >> 4, c = i & 15;
      float v = (r < DD && c < DD) ? o_qc[bt * (DD * DD) + r * DD + c] : 0.0f;
      if (r == c && r < DD) v += 1e-6f;
      MSL[i] = v;
    }
    __syncthreads();
    chol12(MSL, lane);
    if (lane < DD) {
      float s = o_qm[bt * DD + lane];
      for (int j = 0; j <= lane; ++j) s += MSL[lane * 16 + j] * eps_g[bt * DD + j];
      vz[lane] = s;
      o_z[bt * DD + lane] = s;
    }
    __syncthreads();
    if (lane < OBS) {
      float s = sdecb[lane];
      for (int i = 0; i < DD; ++i) s += vz[i] * sdecw[i * 8 + lane];
      o_xr[bt * OBS + lane] = s;
    }
    __syncthreads();
  }
}

// ---------------------------------------------------------------------------
extern "C" void kernel_launch(void* const* d_in, const int* in_sizes, int n_in,
                              void* d_out, int out_size, void* d_ws, size_t ws_size,
                              hipStream_t stream) {
  const float* x      = (const float*)d_in[0];
  const int*   mask   = (const int*)  d_in[1];
  const float* eps    = (const float*)d_in[2];
  const float* enc_w1 = (const float*)d_in[3];
  const float* enc_b1 = (const float*)d_in[4];
  const float* enc_wm = (const float*)d_in[5];
  const float* enc_bm = (const float*)d_in[6];
  const float* enc_wl = (const float*)d_in[7];
  const float* enc_bl = (const float*)d_in[8];
  const float* dec_w  = (const float*)d_in[9];
  const float* dec_b  = (const float*)d_in[10];
  const float* A      = (const float*)d_in[11];
  const float* bvec   = (const float*)d_in[12];
  const float* Qparam = (const float*)d_in[13];

  float* ws    = (float*)d_ws;
  float* xh_ws = ws;                                 // [B,T,D]
  float* sg_ws = xh_ws + (size_t)BB * TT * DD;       // [B,T,D]
  float* Q_ws  = sg_ws + (size_t)BB * TT * DD;       // [D,D]

  float* out  = (float*)d_out;
  float* o_xr = out;                                 // [B,T,OBS]
  float* o_z  = o_xr + (size_t)BB * TT * OBS;        // [B,T,D]
  float* o_qm = o_z  + (size_t)BB * TT * DD;         // [B,T,D]
  float* o_qc = o_qm + (size_t)BB * TT * DD;         // [B,T,D,D]
  float* o_ll = o_qc + (size_t)BB * TT * DD * DD;    // [B]

  svae_enc_kernel<<<(BB * TT + 255) / 256, 256, 0, stream>>>(
      x, enc_w1, enc_b1, enc_wm, enc_bm, enc_wl, enc_bl, xh_ws, sg_ws);

  svae_prep_kernel<<<1, 256, 0, stream>>>(Qparam, Q_ws);

  svae_filter_kernel<<<BB / NW, NW * 32, 0, stream>>>(
      xh_ws, sg_ws, Q_ws, mask, eps, A, bvec, dec_w, dec_b,
      o_xr, o_z, o_qm, o_qc, o_ll);
}